// CenterLoss_3599182594972
// MI455X (gfx1250) — compile-verified
//
#include <hip/hip_runtime.h>

typedef __attribute__((ext_vector_type(2))) float v2f;
typedef __attribute__((ext_vector_type(8))) float v8f;

#define FEAT   2048
#define NCLS   4096
#define BATCH  8192
#define WAVES  8                 // waves per block (256 threads, wave32)
#define KSLICE (FEAT / WAVES)    // 256 K per wave

// One block = one 16-row tile. Each wave covers a 256-wide K slice and
// accumulates diag(diff * diff^T) for its slice with V_WMMA_F32_16X16X4_F32.
// For the symmetric product, the f32 A(16x4) and B(4x16) VGPR layouts coincide
// (lane m holds K={k0,k0+1}, lanes 16-31 hold K={k0+2,k0+3}), so the same
// register pair feeds both A and B.
__global__ __launch_bounds__(256) void center_rowdist_wmma(
    const float* __restrict__ x,
    const int*   __restrict__ labels,
    const float* __restrict__ centers,
    float*       __restrict__ rowdist)
{
    __shared__ float lds[WAVES][16];

    const int tid  = threadIdx.x;
    const int lane = tid & 31;
    const int wave = tid >> 5;
    const int m    = lane & 15;                // row within tile handled by this lane
    const int row  = blockIdx.x * 16 + m;

    const int lab = labels[row];               // 0..4095
    const float* xr = x       + (size_t)row * FEAT;
    const float* cr = centers + (size_t)lab * FEAT;

    const int koff  = (lane < 16) ? 0 : 2;     // K parity per lane half (A/B layout)
    const int kbase = wave * KSLICE + koff;

    v8f acc = {};
    #pragma unroll 4
    for (int kk = 0; kk < KSLICE; kk += 4) {
        const int k = kbase + kk;              // even -> 8B aligned
        v2f xv = *(const v2f*)(xr + k);
        v2f cv = *(const v2f*)(cr + k);
        v2f d;
        d.x = xv.x - cv.x;
        d.y = xv.y - cv.y;
        // D = diff * diff^T + C   (A and B registers identical by symmetry)
        acc = __builtin_amdgcn_wmma_f32_16x16x4_f32(
            /*neg_a=*/false, d, /*neg_b=*/false, d,
            /*c_mod=*/(short)0, acc, /*reuse_a=*/false, /*reuse_b=*/false);
    }

    // Diagonal extraction: D[g][g] -> VGPR g, lane g (g=0..7);
    //                      D[g+8][g+8] -> VGPR g, lane g+24.
    float diag = 0.0f;
    #pragma unroll
    for (int g = 0; g < 8; ++g) {
        float v = acc[g];
        if (lane == g)      diag = v;          // rows 0..7
        if (lane == 24 + g) diag = v;          // rows 8..15
    }
    if (lane < 8)        lds[wave][lane]      = diag;   // row = lane
    else if (lane >= 24) lds[wave][lane - 16] = diag;   // row = lane-16
    __syncthreads();

    // Deterministic fixed-order combine of the 8 K-slice partials per row.
    if (tid < 16) {
        float s = 0.0f;
        #pragma unroll
        for (int w = 0; w < WAVES; ++w) s += lds[w][tid];
        rowdist[blockIdx.x * 16 + tid] = s;
    }
}

// Single-block deterministic reduction of the 8192 clipped distances.
__global__ __launch_bounds__(1024) void center_reduce(
    const float* __restrict__ rowdist,
    float*       __restrict__ out)
{
    __shared__ float lds[1024];
    const int tid = threadIdx.x;

    float s = 0.0f;
    #pragma unroll
    for (int i = 0; i < BATCH / 1024; ++i) {
        float d = rowdist[tid + i * 1024];
        d = fminf(fmaxf(d, 1e-12f), 1e12f);    // clip, matching the reference
        s += d;
    }
    lds[tid] = s;
    __syncthreads();

    #pragma unroll
    for (int off = 512; off > 0; off >>= 1) {
        if (tid < off) lds[tid] += lds[tid + off];
        __syncthreads();
    }

    if (tid == 0) {
        // Masked-out entries: (B*C - B) values of 1e-12, i.e. (C-1)*1e-12 after /B.
        out[0] = lds[0] / (float)BATCH + (float)(NCLS - 1) * 1e-12f;
    }
}

extern "C" void kernel_launch(void* const* d_in, const int* in_sizes, int n_in,
                              void* d_out, int out_size, void* d_ws, size_t ws_size,
                              hipStream_t stream) {
    const float* x       = (const float*)d_in[0];
    const int*   labels  = (const int*)  d_in[1];
    const float* centers = (const float*)d_in[2];
    float* rowdist = (float*)d_ws;             // 8192 floats = 32 KB scratch

    center_rowdist_wmma<<<dim3(BATCH / 16), dim3(256), 0, stream>>>(
        x, labels, centers, rowdist);
    center_reduce<<<dim3(1), dim3(1024), 0, stream>>>(
        rowdist, (float*)d_out);
}